// Integration_block_29781303230698
// MI455X (gfx1250) — compile-verified
//
#include <hip/hip_runtime.h>

// Integration of stationary velocity field by scaling-and-squaring.
// SHAPE = (1, 3, 160, 192, 224), INT_STEPS = 7.
// flow_{k+1}(x) = flow_k(x) + trilerp(flow_k, x + flow_k(x)), zero padding.
//
// Memory plan: ping-pong between d_out and d_ws (each 3*N floats, 82.6 MB).
// Both buffers (165 MB) fit in the 192 MB global L2, so steps 2..7 run out
// of L2. Step 1 reads SVF from HBM and folds the 1/128 scaling into the
// interpolation weights.
//
// Launch geometry: blockDim.x = W = 224 (7 wave32s), grid = (1, H, D).
// This removes all integer div/mod from the index decomposition (the old
// version burned ~12 VALU ops in v_mul_hi_i32 magic-number division).

#define DD 160
#define HH 192
#define WW 224
#define HWSZ (HH * WW)          // 43008
#define NVOX (DD * HH * WW)     // 6881280

template <bool FIRST>
__global__ __launch_bounds__(WW)
void integ_step(const float* __restrict__ src, float* __restrict__ dst) {
    const int w = threadIdx.x;   // 0..223
    const int h = blockIdx.y;    // 0..191
    const int d = blockIdx.z;    // 0..159
    const int x = (d * HH + h) * WW + w;

    const float s = FIRST ? (1.0f / 128.0f) : 1.0f;

    // Coalesced center reads (channel-planar layout).
    const float fd = s * src[x];
    const float fh = s * src[NVOX + x];
    const float fw = s * src[2 * NVOX + x];

    // Sample position in voxel coordinates (align_corners=True voxel-space
    // sampling, per reference).
    const float pd = (float)d + fd;
    const float ph = (float)h + fh;
    const float pw = (float)w + fw;

    const float fld = floorf(pd), flh = floorf(ph), flw = floorf(pw);
    const int d0 = (int)fld, h0 = (int)flh, w0 = (int)flw;

    float td = pd - fld, th = ph - flh, tw = pw - flw;
    float sd = 1.0f - td, sh = 1.0f - th, sw = 1.0f - tw;
    // Every trilinear weight has exactly one d-axis factor: fold the
    // FIRST-step 1/128 into (sd, td) so all 8 weights carry it once.
    sd *= s; td *= s;

    float acc0 = 0.0f, acc1 = 0.0f, acc2 = 0.0f;

    const bool interior = ((unsigned)d0 < (unsigned)(DD - 1)) &
                          ((unsigned)h0 < (unsigned)(HH - 1)) &
                          ((unsigned)w0 < (unsigned)(WW - 1));
    if (interior) {
        const int base = (d0 * HH + h0) * WW + w0;
        // gfx1250 global_prefetch_b8: warm the gather cachelines of the far
        // channel planes while the weight ALU work is in flight.
        __builtin_prefetch(src + NVOX + base, 0, 3);
        __builtin_prefetch(src + 2 * NVOX + base, 0, 3);

        const float w000 = sd * sh * sw, w001 = sd * sh * tw;
        const float w010 = sd * th * sw, w011 = sd * th * tw;
        const float w100 = td * sh * sw, w101 = td * sh * tw;
        const float w110 = td * th * sw, w111 = td * th * tw;

        {
            const float* p = src + base;
            acc0 = w000 * p[0]         + w001 * p[1]
                 + w010 * p[WW]        + w011 * p[WW + 1]
                 + w100 * p[HWSZ]      + w101 * p[HWSZ + 1]
                 + w110 * p[HWSZ + WW] + w111 * p[HWSZ + WW + 1];
        }
        {
            const float* p = src + NVOX + base;
            acc1 = w000 * p[0]         + w001 * p[1]
                 + w010 * p[WW]        + w011 * p[WW + 1]
                 + w100 * p[HWSZ]      + w101 * p[HWSZ + 1]
                 + w110 * p[HWSZ + WW] + w111 * p[HWSZ + WW + 1];
        }
        {
            const float* p = src + 2 * NVOX + base;
            acc2 = w000 * p[0]         + w001 * p[1]
                 + w010 * p[WW]        + w011 * p[WW + 1]
                 + w100 * p[HWSZ]      + w101 * p[HWSZ + 1]
                 + w110 * p[HWSZ + WW] + w111 * p[HWSZ + WW + 1];
        }
    } else {
        // Boundary shell: per-corner validity, zero padding.
        #pragma unroll
        for (int i = 0; i < 2; ++i) {
            const int dd = d0 + i;
            const float wd = i ? td : sd;
            #pragma unroll
            for (int j = 0; j < 2; ++j) {
                const int hh2 = h0 + j;
                const float wh2 = j ? th : sh;
                #pragma unroll
                for (int k = 0; k < 2; ++k) {
                    const int ww2 = w0 + k;
                    const float wt = wd * wh2 * (k ? tw : sw);
                    const bool ok = ((unsigned)dd < (unsigned)DD) &
                                    ((unsigned)hh2 < (unsigned)HH) &
                                    ((unsigned)ww2 < (unsigned)WW);
                    if (ok) {
                        const int idx = (dd * HH + hh2) * WW + ww2;
                        acc0 += wt * src[idx];
                        acc1 += wt * src[NVOX + idx];
                        acc2 += wt * src[2 * NVOX + idx];
                    }
                }
            }
        }
    }

    dst[x]            = fd + acc0;
    dst[NVOX + x]     = fh + acc1;
    dst[2 * NVOX + x] = fw + acc2;
}

extern "C" void kernel_launch(void* const* d_in, const int* in_sizes, int n_in,
                              void* d_out, int out_size, void* d_ws, size_t ws_size,
                              hipStream_t stream) {
    (void)in_sizes; (void)n_in; (void)out_size; (void)ws_size;
    const float* svf = (const float*)d_in[0];
    float* out = (float*)d_out;
    float* ws  = (float*)d_ws;   // needs 3*NVOX floats = 82.6 MB scratch

    const dim3 blk(WW, 1, 1);    // 224 threads = 7 wave32s, one W-row
    const dim3 grd(1, HH, DD);   // 192 x 160 rows

    // 7 steps (odd count): start SVF -> out, ping-pong, end in out.
    integ_step<true ><<<grd, blk, 0, stream>>>(svf, out);
    integ_step<false><<<grd, blk, 0, stream>>>(out, ws);
    integ_step<false><<<grd, blk, 0, stream>>>(ws,  out);
    integ_step<false><<<grd, blk, 0, stream>>>(out, ws);
    integ_step<false><<<grd, blk, 0, stream>>>(ws,  out);
    integ_step<false><<<grd, blk, 0, stream>>>(out, ws);
    integ_step<false><<<grd, blk, 0, stream>>>(ws,  out);
}